// GCN_11871289606692
// MI455X (gfx1250) — compile-verified
//
#include <hip/hip_runtime.h>
#include <hip/hip_bf16.h>

#define N_NODES 50000
#define N_EDGES 1600000
#define D_IN 128
#define D_HID 128
#define D_OUT 64
#define N_EXPERT 16

typedef float v2f __attribute__((ext_vector_type(2)));
typedef float v8f __attribute__((ext_vector_type(8)));

// ---------------- reset counters ----------------
__global__ void k_reset(int* count, int* cursor, float* gate_sum) {
  int t = threadIdx.x;
  if (t < N_EXPERT) { count[t] = 0; cursor[t] = 0; }
  if (gate_sum != nullptr && t < 2) gate_sum[t] = 0.0f;
}

// ---------------- gating: softmax std + argmax expert ----------------
// one wave (32 lanes) per node; 8 nodes per 256-thread block
__global__ void k_gate(const float* __restrict__ X, const float* __restrict__ Wg,
                       const float* __restrict__ bg, int* __restrict__ node_expert,
                       int* __restrict__ count, float* __restrict__ gate_sum) {
  __shared__ float WgS[128 * 17];  // stride 17 -> conflict-free column reads
  int tid = threadIdx.x;
  for (int i = tid; i < 128 * N_EXPERT; i += 256) {
    int k = i >> 4, e = i & 15;
    WgS[k * 17 + e] = Wg[i];
  }
  __syncthreads();
  int node = blockIdx.x * 8 + (tid >> 5);
  if (node >= N_NODES) return;
  int lane = tid & 31;

  float acc[N_EXPERT];
#pragma unroll
  for (int e = 0; e < N_EXPERT; ++e) acc[e] = 0.0f;
  const float* xr = X + (size_t)node * D_IN;
#pragma unroll
  for (int j = 0; j < 4; ++j) {
    int k = lane + 32 * j;
    float xv = xr[k];
#pragma unroll
    for (int e = 0; e < N_EXPERT; ++e) acc[e] += xv * WgS[k * 17 + e];
  }
  // wave32 xor-reduction
#pragma unroll
  for (int m = 16; m >= 1; m >>= 1) {
#pragma unroll
    for (int e = 0; e < N_EXPERT; ++e) acc[e] += __shfl_xor(acc[e], m, 32);
  }
  if (lane == 0) {
    float logits[N_EXPERT];
    float mx = -1e30f;
    int best = 0;
#pragma unroll
    for (int e = 0; e < N_EXPERT; ++e) {
      logits[e] = acc[e] + bg[e];
      if (logits[e] > mx) { mx = logits[e]; best = e; }  // first-occurrence argmax
    }
    float p[N_EXPERT];
    float s = 0.0f;
#pragma unroll
    for (int e = 0; e < N_EXPERT; ++e) { p[e] = expf(logits[e] - mx); s += p[e]; }
    float inv = 1.0f / s;
    float mean = 0.0f;
#pragma unroll
    for (int e = 0; e < N_EXPERT; ++e) { p[e] *= inv; mean += p[e]; }
    mean *= (1.0f / 16.0f);
    float var = 0.0f;
#pragma unroll
    for (int e = 0; e < N_EXPERT; ++e) { float d = p[e] - mean; var += d * d; }
    var *= (1.0f / 15.0f);  // ddof = 1
    atomicAdd(gate_sum, sqrtf(var));
    node_expert[node] = best;
    atomicAdd(&count[best], 1);
  }
}

// ---------------- 16-entry exclusive scan ----------------
__global__ void k_scan(const int* __restrict__ count, int* __restrict__ offset) {
  if (threadIdx.x == 0 && blockIdx.x == 0) {
    int s = 0;
    for (int e = 0; e < N_EXPERT; ++e) { offset[e] = s; s += count[e]; }
    offset[N_EXPERT] = s;
  }
}

// ---------------- bucket nodes by expert ----------------
__global__ void k_bucket(const int* __restrict__ node_expert, const int* __restrict__ offset,
                         int* __restrict__ cursor, int* __restrict__ bucket) {
  int n = blockIdx.x * 256 + threadIdx.x;
  if (n >= N_NODES) return;
  int e = node_expert[n];
  int pos = atomicAdd(&cursor[e], 1);
  bucket[offset[e] + pos] = n;
}

// ---------------- MoE expert GEMM: 16-node tile, one expert, WMMA fp32 ----------
// block: 256 threads = 8 waves; wave w owns N-columns [16w, 16w+16); K = 128
__global__ void k_moe_gemm(const float* __restrict__ X, const float* __restrict__ We,
                           const float* __restrict__ be, const int* __restrict__ bucket,
                           const int* __restrict__ offset, const int* __restrict__ count,
                           float* __restrict__ H) {
  __shared__ float xt[16 * 130];  // padded stride
  __shared__ int s_node[16];

  // map blockIdx -> (expert, tile); uniform across the whole block
  int t = blockIdx.x;
  int expert = -1, tile = 0;
  for (int e = 0; e < N_EXPERT; ++e) {
    int ct = (count[e] + 15) >> 4;
    if (t < ct) { expert = e; tile = t; break; }
    t -= ct;
  }
  if (expert < 0) return;
  int base = offset[expert] + tile * 16;
  int valid = count[expert] - tile * 16;
  if (valid > 16) valid = 16;

  int tid = threadIdx.x;
  if (tid < 16) {
    int r = tid < valid ? tid : (valid - 1);  // pad with a valid row (never stored)
    s_node[tid] = bucket[base + r];
  }
  __syncthreads();
  for (int i = tid; i < 16 * 128; i += 256) {
    int r = i >> 7, c = i & 127;
    xt[r * 130 + c] = X[(size_t)s_node[r] * D_HID + c];
  }
  __syncthreads();

  int wave = tid >> 5, lane = tid & 31;
  int half = lane >> 4, ln = lane & 15;
  int nb = wave * 16;
  const float* W = We + (size_t)expert * D_HID * D_HID;

  v8f c = {};
  for (int k0 = 0; k0 < 128; k0 += 4) {
    int ka = k0 + 2 * half;
    v2f a, b;
    a.x = xt[ln * 130 + ka];
    a.y = xt[ln * 130 + ka + 1];
    b.x = W[(size_t)ka * D_HID + nb + ln];
    b.y = W[(size_t)(ka + 1) * D_HID + nb + ln];
    c = __builtin_amdgcn_wmma_f32_16x16x4_f32(false, a, false, b, (short)0, c, false, false);
  }
  float bias = be[expert * D_HID + nb + ln];
#pragma unroll
  for (int r = 0; r < 8; ++r) {
    int m = r + 8 * half;
    if (m < valid) H[(size_t)s_node[m] * D_HID + nb + ln] = c[r] + bias;
  }
}

// ---------------- plain final GEMM 128 -> 64 (no bucketing; 50000 = 3125*16) ----
// block: 128 threads = 4 waves
__global__ void k_gemm_last(const float* __restrict__ X, const float* __restrict__ W,
                            float* __restrict__ H) {
  __shared__ float xt[16 * 130];
  int tid = threadIdx.x;
  int row0 = blockIdx.x * 16;
  for (int i = tid; i < 16 * 128; i += 128) {
    int r = i >> 7, cc = i & 127;
    xt[r * 130 + cc] = X[(size_t)(row0 + r) * D_HID + cc];
  }
  __syncthreads();
  int wave = tid >> 5, lane = tid & 31;
  int half = lane >> 4, ln = lane & 15;
  int nb = wave * 16;
  v8f c = {};
  for (int k0 = 0; k0 < 128; k0 += 4) {
    int ka = k0 + 2 * half;
    v2f a, b;
    a.x = xt[ln * 130 + ka];
    a.y = xt[ln * 130 + ka + 1];
    b.x = W[(size_t)ka * D_OUT + nb + ln];
    b.y = W[(size_t)(ka + 1) * D_OUT + nb + ln];
    c = __builtin_amdgcn_wmma_f32_16x16x4_f32(false, a, false, b, (short)0, c, false, false);
  }
#pragma unroll
  for (int r = 0; r < 8; ++r) {
    int m = r + 8 * half;
    H[(size_t)(row0 + m) * D_OUT + nb + ln] = c[r];
  }
}

// ---------------- init buffer with broadcast bias ----------------
__global__ void k_init_bias(float* __restrict__ buf, const float* __restrict__ bias, int cmask) {
  int i = blockIdx.x * 256 + threadIdx.x;
  buf[i] = bias[i & cmask];
}

// ---------------- edge scatter-add (segment sum), 4 channels / thread ----------
__global__ void k_scatter(const float* __restrict__ h, float* __restrict__ out,
                          const int* __restrict__ src, const int* __restrict__ dst,
                          int log2cpt /* log2(C/4) */, int C) {
  long long tid = (long long)blockIdx.x * 256 + threadIdx.x;
  int e = (int)(tid >> log2cpt);
  if (e >= N_EDGES) return;
  int cb = ((int)tid & ((1 << log2cpt) - 1)) << 2;
  const float4 v = *(const float4*)(h + (size_t)src[e] * C + cb);
  float* o = out + (size_t)dst[e] * C + cb;
  atomicAdd(o + 0, v.x);
  atomicAdd(o + 1, v.y);
  atomicAdd(o + 2, v.z);
  atomicAdd(o + 3, v.w);
}

// ---------------- relu ----------------
__global__ void k_relu(float* __restrict__ buf) {
  int i = blockIdx.x * 256 + threadIdx.x;
  buf[i] = fmaxf(buf[i], 0.0f);
}

// ---------------- final scalars ----------------
__global__ void k_finalize(float* __restrict__ out_tail, const float* __restrict__ gate_sum) {
  if (threadIdx.x == 0 && blockIdx.x == 0) {
    out_tail[0] = 0.5f * (gate_sum[0] + gate_sum[1]) * (1.0f / (float)N_NODES);
    out_tail[1] = 1.0f;
  }
}

extern "C" void kernel_launch(void* const* d_in, const int* in_sizes, int n_in,
                              void* d_out, int out_size, void* d_ws, size_t ws_size,
                              hipStream_t stream) {
  const float* x     = (const float*)d_in[0];
  const int*   edge  = (const int*)d_in[1];   // [2, N_EDGES]
  const float* Wg1   = (const float*)d_in[2];
  const float* bg1   = (const float*)d_in[3];
  const float* We1   = (const float*)d_in[4];
  const float* be1   = (const float*)d_in[5];
  const float* b1    = (const float*)d_in[6];
  const float* Wg2   = (const float*)d_in[7];
  const float* bg2   = (const float*)d_in[8];
  const float* We2   = (const float*)d_in[9];
  const float* be2   = (const float*)d_in[10];
  const float* b2    = (const float*)d_in[11];
  const float* Wlast = (const float*)d_in[12];
  const float* blast = (const float*)d_in[13];
  const int* esrc = edge;
  const int* edst = edge + N_EDGES;

  // workspace carve-up (256B aligned)
  char* ws = (char*)d_ws;
  size_t off = 0;
  auto carve = [&](size_t bytes) {
    char* p = ws + off;
    off = (off + bytes + 255) & ~(size_t)255;
    return p;
  };
  float* bufA = (float*)carve((size_t)N_NODES * D_HID * sizeof(float));  // pre-scatter h
  float* bufB = (float*)carve((size_t)N_NODES * D_HID * sizeof(float));  // activations
  float* bufC = (float*)carve((size_t)N_NODES * D_OUT * sizeof(float));  // h3
  int* node_expert = (int*)carve((size_t)N_NODES * sizeof(int));
  int* bucket      = (int*)carve((size_t)N_NODES * sizeof(int));
  int* count  = (int*)carve(N_EXPERT * sizeof(int));
  int* offset = (int*)carve((N_EXPERT + 1) * sizeof(int));
  int* cursor = (int*)carve(N_EXPERT * sizeof(int));
  float* gate_sum = (float*)carve(2 * sizeof(float));

  const int gate_blocks = (N_NODES + 7) / 8;                       // 6250
  const int bucket_blocks = (N_NODES + 255) / 256;                 // 196
  const int moe_blocks = N_NODES / 16 + N_EXPERT;                  // 3141 (worst-case tiles)
  const int init128_blocks = N_NODES * D_HID / 256;                // 25000
  const int init64_blocks = N_NODES * D_OUT / 256;                 // 12500
  const int scat128_blocks = (int)((long long)N_EDGES * 32 / 256); // 200000
  const int scat64_blocks = (int)((long long)N_EDGES * 16 / 256);  // 100000

  // ---- layer 1 ----
  k_reset<<<1, 64, 0, stream>>>(count, cursor, gate_sum);
  k_gate<<<gate_blocks, 256, 0, stream>>>(x, Wg1, bg1, node_expert, count, gate_sum + 0);
  k_scan<<<1, 32, 0, stream>>>(count, offset);
  k_bucket<<<bucket_blocks, 256, 0, stream>>>(node_expert, offset, cursor, bucket);
  k_moe_gemm<<<moe_blocks, 256, 0, stream>>>(x, We1, be1, bucket, offset, count, bufA);
  k_init_bias<<<init128_blocks, 256, 0, stream>>>(bufB, b1, D_HID - 1);
  k_scatter<<<scat128_blocks, 256, 0, stream>>>(bufA, bufB, esrc, edst, 5, D_HID);
  k_relu<<<init128_blocks, 256, 0, stream>>>(bufB);

  // ---- layer 2 ----
  k_reset<<<1, 64, 0, stream>>>(count, cursor, nullptr);
  k_gate<<<gate_blocks, 256, 0, stream>>>(bufB, Wg2, bg2, node_expert, count, gate_sum + 1);
  k_scan<<<1, 32, 0, stream>>>(count, offset);
  k_bucket<<<bucket_blocks, 256, 0, stream>>>(node_expert, offset, cursor, bucket);
  k_moe_gemm<<<moe_blocks, 256, 0, stream>>>(bufB, We2, be2, bucket, offset, count, bufA);
  k_init_bias<<<init128_blocks, 256, 0, stream>>>(bufB, b2, D_HID - 1);
  k_scatter<<<scat128_blocks, 256, 0, stream>>>(bufA, bufB, esrc, edst, 5, D_HID);
  k_relu<<<init128_blocks, 256, 0, stream>>>(bufB);

  // ---- final plain GCN layer ----
  float* out = (float*)d_out;  // [N_NODES*D_OUT] + 2 scalars
  k_gemm_last<<<N_NODES / 16, 128, 0, stream>>>(bufB, Wlast, bufC);
  k_init_bias<<<init64_blocks, 256, 0, stream>>>(out, blast, D_OUT - 1);
  k_scatter<<<scat64_blocks, 256, 0, stream>>>(bufC, out, esrc, edst, 4, D_OUT);
  k_finalize<<<1, 32, 0, stream>>>(out + (size_t)N_NODES * D_OUT, gate_sum);
}